// RPNPostProcessor_63479616635111
// MI455X (gfx1250) — compile-verified
//
#include <hip/hip_runtime.h>
#include <hip/hip_bf16.h>
#include <stdint.h>

// ---------------------------------------------------------------------------
// RPN post-processor for MI455X (gfx1250, wave32).
//   1) score_hist_kernel : sigmoid(objectness) -> (h,w,a)-ordered score plane
//                          FUSED with radix pass-0 (MSB byte) LDS histogram
//   2) 3x (hist, select) : exact radix-select of the 1000th-largest score key
//                          per image (sigmoid > 0 => raw IEEE bits order ==
//                          float order). select_kernel re-zeroes hist bins.
//   3) compact_kernel    : gather (key,idx) with key >= threshold (<=1024)
//   4) sort_kernel       : gfx1250 async global->LDS staging
//                          (global_load_async_to_lds_b64 + s_wait_asynccnt)
//                          + LDS bitonic sort, key desc, index tiebreak
//   5) decode_kernel     : box delta decode + clip (reference math)
//   6) mask_kernel       : 1000x1000 IoU suppression bits (j > i only)
//   7) nms_reduce_kernel : one wave32 per image; lane l owns suppression
//                          word l; sequential greedy NMS + shfl-scan ranks
//   8) output_kernel     : survivors (desc score), then (0,0,0,0,NEG) rows
// No WMMA: no matmul structure exists in this op (IoU = min/max/mul).
// gfx1250 paths: async-to-LDS, s_wait_asynccnt, global_prefetch_b8, wave32.
// ---------------------------------------------------------------------------

#define NIMG  16
#define ANCH  3
#define HH    200
#define WW    336
#define HWP   (HH * WW)          // 67200
#define HWA   (HWP * ANCH)       // 201600
#define TOPK  1000
#define CAP   1024
#define NEGV  (-10000.0f)
#define NMS_T 0.7f
#define IMGWF 1344.0f
#define IMGHF 800.0f
#define XCLIP 4.135166556742356f // log(1000/16)
#define CHUNKS ((HWA + 1023) / 1024)   // 197

// ---------------- init: zero hist + per-image scalars -----------------------
__global__ void init_kernel(unsigned* hist, unsigned* prefix, int* krem,
                            int* ccnt, int* count) {
  int img = blockIdx.x;
  hist[img * 256 + threadIdx.x] = 0u;
  if (threadIdx.x == 0) {
    prefix[img] = 0u;
    krem[img]   = TOPK;
    ccnt[img]   = 0;
    count[img]  = 0;
  }
}

// ---------------- 1) sigmoid + transpose, fused radix pass 0 ----------------
__global__ void score_hist_kernel(const float* __restrict__ obj,
                                  float* __restrict__ scores,
                                  unsigned* __restrict__ hist) {
  __shared__ unsigned lh[256];
  int img   = blockIdx.x / CHUNKS;
  int chunk = blockIdx.x - img * CHUNKS;
  lh[threadIdx.x] = 0u;
  __syncthreads();
  int base = chunk * 1024;
  for (int t = threadIdx.x; t < 1024; t += blockDim.x) {
    int j = base + t;
    if (j < HWA) {
      int a  = j % ANCH;
      int hw = j / ANCH;
      int h  = hw / WW;
      int w  = hw - h * WW;
      float x = obj[((img * ANCH + a) * HH + h) * WW + w];
      float s = 1.0f / (1.0f + expf(-x));
      scores[img * HWA + j] = s;
      atomicAdd(&lh[__float_as_uint(s) >> 24], 1u);   // MSB-byte histogram
    }
  }
  __syncthreads();
  unsigned c = lh[threadIdx.x];
  if (c) atomicAdd(&hist[img * 256 + threadIdx.x], c);
}

// ---------------- 2) radix passes 1..3 histogram ----------------------------
__global__ void hist_kernel(const float* __restrict__ scores,
                            unsigned* __restrict__ hist,
                            const unsigned* __restrict__ prefix,
                            unsigned shift, unsigned highmask) {
  __shared__ unsigned lh[256];
  int img   = blockIdx.x / CHUNKS;
  int chunk = blockIdx.x - img * CHUNKS;
  lh[threadIdx.x] = 0u;
  __syncthreads();
  unsigned pre  = prefix[img];
  int      base = chunk * 1024;
  const float* sp = scores + img * HWA;
  for (int t = threadIdx.x; t < 1024; t += blockDim.x) {
    int j = base + t;
    if (j < HWA) {
      __builtin_prefetch(sp + j + 256, 0, 1);   // -> global_prefetch_b8
      unsigned key = __float_as_uint(sp[j]);
      if ((key & highmask) == pre)
        atomicAdd(&lh[(key >> shift) & 255u], 1u);
    }
  }
  __syncthreads();
  unsigned c = lh[threadIdx.x];
  if (c) atomicAdd(&hist[img * 256 + threadIdx.x], c);
}

// select digit (scan bins high->low), then re-zero this image's bins ---------
__global__ void select_kernel(unsigned* __restrict__ hist,
                              unsigned* __restrict__ prefix, int* __restrict__ krem,
                              unsigned shift) {
  int img = threadIdx.x;
  if (img >= NIMG) return;
  int need = krem[img];
  int cum  = 0;
  int d    = 255;
  for (; d > 0; --d) {
    int c = (int)hist[img * 256 + d];
    if (cum + c >= need) break;
    cum += c;
  }
  prefix[img] |= ((unsigned)d) << shift;
  krem[img]    = need - cum;
  for (int b = 0; b < 256; ++b) hist[img * 256 + b] = 0u;  // ready for next pass
}

// ---------------- 3) candidate compaction -----------------------------------
__global__ void compact_kernel(const float* __restrict__ scores,
                               const unsigned* __restrict__ prefix,
                               unsigned* __restrict__ ckey, int* __restrict__ cidx,
                               int* __restrict__ ccnt) {
  int t = blockIdx.x * blockDim.x + threadIdx.x;
  if (t >= NIMG * HWA) return;
  int img = t / HWA;
  int j   = t - img * HWA;
  unsigned key = __float_as_uint(scores[t]);
  if (key >= prefix[img]) {
    int pos = atomicAdd(&ccnt[img], 1);
    if (pos < CAP) {
      ckey[img * CAP + pos] = key;
      cidx[img * CAP + pos] = j;
    }
  }
}

// ---------------- 4) bitonic sort in LDS (async global->LDS staging) --------
__global__ __launch_bounds__(512) void sort_kernel(
    const unsigned* __restrict__ ckey, const int* __restrict__ cidx,
    const int* __restrict__ ccnt,
    int* __restrict__ top_idx, float* __restrict__ top_score) {
  __shared__ unsigned           skey[CAP];
  __shared__ int                sidx[CAP];
  __shared__ unsigned long long items[CAP];
  int img = blockIdx.x;
  int tid = threadIdx.x;               // 512 threads, 2 slots each

  // gfx1250 async copy: global -> LDS, tracked by ASYNCcnt.
  // Low 32 bits of the generic address of a __shared__ object are its
  // wave-relative LDS byte offset (flat->LDS mapping discards upper bits).
  {
    unsigned lds_k = (unsigned)(unsigned long long)(&skey[2 * tid]);
    unsigned lds_i = (unsigned)(unsigned long long)(&sidx[2 * tid]);
    const unsigned* gk = &ckey[img * CAP + 2 * tid];
    const int*      gi = &cidx[img * CAP + 2 * tid];
    asm volatile("global_load_async_to_lds_b64 %0, %1, off"
                 :: "v"(lds_k), "v"(gk) : "memory");
    asm volatile("global_load_async_to_lds_b64 %0, %1, off"
                 :: "v"(lds_i), "v"(gi) : "memory");
    asm volatile("s_wait_asynccnt 0" ::: "memory");
  }
  __syncthreads();

  int cnt = ccnt[img];
  if (cnt > CAP) cnt = CAP;
  for (int s = tid; s < CAP; s += 512) {
    unsigned k = (s < cnt) ? skey[s] : 0u;
    unsigned j = (s < cnt) ? (unsigned)sidx[s] : 0x7FFFFFFFu;
    // key descending; for equal keys, smaller index first (~j descending)
    items[s] = ((unsigned long long)k << 32) | (unsigned)(~j);
  }
  __syncthreads();

  for (unsigned k = 2; k <= CAP; k <<= 1) {
    for (unsigned j = k >> 1; j > 0; j >>= 1) {
      unsigned i = ((tid & ~(j - 1)) << 1) | (tid & (j - 1));
      unsigned p = i | j;
      bool desc = ((i & k) == 0);
      unsigned long long a = items[i], b = items[p];
      if (desc ? (a < b) : (a > b)) { items[i] = b; items[p] = a; }
      __syncthreads();
    }
  }
  for (int r = tid; r < TOPK; r += 512) {
    unsigned long long it = items[r];
    top_idx[img * TOPK + r]   = (int)(~(unsigned)it);
    top_score[img * TOPK + r] = __uint_as_float((unsigned)(it >> 32));
  }
}

// ---------------- 5) box decode + clip --------------------------------------
__global__ void decode_kernel(const float* __restrict__ box_reg,
                              const float* __restrict__ anchors,
                              const int* __restrict__ top_idx,
                              float* __restrict__ boxes) {
  int t = blockIdx.x * blockDim.x + threadIdx.x;
  if (t >= NIMG * TOPK) return;
  int img = t / TOPK, r = t - img * TOPK;
  int j  = top_idx[img * TOPK + r];
  int a  = j % ANCH;
  int hw = j / ANCH;
  int h  = hw / WW;
  int w  = hw - h * WW;
  const float* br = box_reg + (size_t)img * (ANCH * 4 * HWP);
  int base = a * 4 * HWP + h * WW + w;   // box_regression[n, a, c, h, w]
  float dx = br[base];
  float dy = br[base + HWP];
  float dw = fminf(br[base + 2 * HWP], XCLIP);
  float dh = fminf(br[base + 3 * HWP], XCLIP);
  const float* an = anchors + ((size_t)img * HWA + j) * 4;
  float aw = an[2] - an[0] + 1.0f;
  float ah = an[3] - an[1] + 1.0f;
  float cx = an[0] + 0.5f * aw;
  float cy = an[1] + 0.5f * ah;
  float pcx = dx * aw + cx, pcy = dy * ah + cy;
  float pw  = expf(dw) * aw, ph = expf(dh) * ah;
  float x1 = fminf(fmaxf(pcx - 0.5f * pw, 0.0f), IMGWF - 1.0f);
  float y1 = fminf(fmaxf(pcy - 0.5f * ph, 0.0f), IMGHF - 1.0f);
  float x2 = fminf(fmaxf(pcx + 0.5f * pw - 1.0f, 0.0f), IMGWF - 1.0f);
  float y2 = fminf(fmaxf(pcy + 0.5f * ph - 1.0f, 0.0f), IMGHF - 1.0f);
  float* o = boxes + (size_t)(img * TOPK + r) * 4;
  o[0] = x1; o[1] = y1; o[2] = x2; o[3] = y2;
}

// ---------------- 6) pairwise IoU suppression bits --------------------------
__global__ void mask_kernel(const float* __restrict__ boxes, unsigned* __restrict__ mask) {
  int t = blockIdx.x * blockDim.x + threadIdx.x;
  if (t >= NIMG * TOPK * 32) return;
  int img = t / (TOPK * 32);
  int rem = t - img * (TOPK * 32);
  int i   = rem >> 5;
  int ww  = rem & 31;
  const float* bi = boxes + (size_t)(img * TOPK + i) * 4;
  float ix1 = bi[0], iy1 = bi[1], ix2 = bi[2], iy2 = bi[3];
  float ai = (ix2 - ix1 + 1.0f) * (iy2 - iy1 + 1.0f);
  unsigned bits = 0u;
  int j0 = ww << 5;
  for (int b = 0; b < 32; ++b) {
    int j = j0 + b;
    if (j >= TOPK || j <= i) continue;
    const float* bj = boxes + (size_t)(img * TOPK + j) * 4;
    float xx1 = fmaxf(ix1, bj[0]);
    float yy1 = fmaxf(iy1, bj[1]);
    float xx2 = fminf(ix2, bj[2]);
    float yy2 = fminf(iy2, bj[3]);
    float inter = fmaxf(xx2 - xx1 + 1.0f, 0.0f) * fmaxf(yy2 - yy1 + 1.0f, 0.0f);
    float aj = (bj[2] - bj[0] + 1.0f) * (bj[3] - bj[1] + 1.0f);
    float iou = inter / (ai + aj - inter);
    if (iou > NMS_T) bits |= (1u << b);
  }
  mask[(size_t)(img * TOPK + i) * 32 + ww] = bits;
}

// ---------------- 7) greedy NMS reduce: one wave32 per image ----------------
__global__ void nms_reduce_kernel(const unsigned* __restrict__ mask,
                                  int* __restrict__ ord, int* __restrict__ count) {
  int img  = blockIdx.x;
  int lane = threadIdx.x;              // 0..31, lane l owns bits [32l, 32l+32)
  unsigned removed = 0u;
  for (int i = 0; i < TOPK; ++i) {
    int owner = i >> 5;
    unsigned rw = (unsigned)__shfl((int)removed, owner, 32);
    if (!((rw >> (i & 31)) & 1u))
      removed |= mask[(size_t)(img * TOPK + i) * 32 + lane];
  }
  unsigned validmask = (lane < 31) ? 0xFFFFFFFFu : 0x000000FFu; // 1000 = 31*32+8
  unsigned kept = (~removed) & validmask;
  int cnt  = __popc(kept);
  int scan = cnt;
  for (int d = 1; d < 32; d <<= 1) {
    int v = __shfl_up(scan, (unsigned)d, 32);
    if (lane >= d) scan += v;
  }
  int pos = scan - cnt;                // exclusive prefix
  for (int b = 0; b < 32; ++b)
    if ((kept >> b) & 1u) ord[img * TOPK + (pos++)] = (lane << 5) + b;
  if (lane == 31) count[img] = scan;   // inclusive at last lane == total kept
}

// ---------------- 8) final output (N, 1000, 5) ------------------------------
__global__ void output_kernel(const float* __restrict__ boxes,
                              const float* __restrict__ top_score,
                              const int* __restrict__ ord, const int* __restrict__ count,
                              float* __restrict__ out) {
  int t = blockIdx.x * blockDim.x + threadIdx.x;
  if (t >= NIMG * TOPK) return;
  int img = t / TOPK, r = t - img * TOPK;
  float* o = out + (size_t)(img * TOPK + r) * 5;
  if (r < count[img]) {
    int i = ord[img * TOPK + r];
    const float* b = boxes + (size_t)(img * TOPK + i) * 4;
    o[0] = b[0]; o[1] = b[1]; o[2] = b[2]; o[3] = b[3];
    o[4] = top_score[img * TOPK + i];
  } else {
    o[0] = 0.0f; o[1] = 0.0f; o[2] = 0.0f; o[3] = 0.0f; o[4] = NEGV;
  }
}

// ---------------------------------------------------------------------------
extern "C" void kernel_launch(void* const* d_in, const int* in_sizes, int n_in,
                              void* d_out, int out_size, void* d_ws, size_t ws_size,
                              hipStream_t stream) {
  (void)in_sizes; (void)n_in; (void)out_size; (void)ws_size;
  const float* objectness = (const float*)d_in[0];
  const float* box_reg    = (const float*)d_in[1];
  const float* anchors    = (const float*)d_in[2];
  float*       out        = (float*)d_out;

  // Workspace carve-out (~15.5 MB total), 256B aligned sections.
  char* p = (char*)d_ws;
  auto carve = [&p](size_t bytes) -> char* {
    char* q = p;
    p += (bytes + 255) & ~(size_t)255;
    return q;
  };
  float*    scores    = (float*)carve((size_t)NIMG * HWA * 4);        // 12.9 MB
  unsigned* hist      = (unsigned*)carve((size_t)NIMG * 256 * 4);
  unsigned* prefix    = (unsigned*)carve((size_t)NIMG * 4);
  int*      krem      = (int*)carve((size_t)NIMG * 4);
  int*      ccnt      = (int*)carve((size_t)NIMG * 4);
  unsigned* ckey      = (unsigned*)carve((size_t)NIMG * CAP * 4);
  int*      cidx      = (int*)carve((size_t)NIMG * CAP * 4);
  int*      top_idx   = (int*)carve((size_t)NIMG * TOPK * 4);
  float*    top_score = (float*)carve((size_t)NIMG * TOPK * 4);
  float*    boxes     = (float*)carve((size_t)NIMG * TOPK * 4 * 4);
  unsigned* supmask   = (unsigned*)carve((size_t)NIMG * TOPK * 32 * 4); // 2 MB
  int*      ord       = (int*)carve((size_t)NIMG * TOPK * 4);
  int*      count     = (int*)carve((size_t)NIMG * 4);

  const int total = NIMG * HWA;                  // 3,225,600

  init_kernel<<<NIMG, 256, 0, stream>>>(hist, prefix, krem, ccnt, count);
  score_hist_kernel<<<NIMG * CHUNKS, 256, 0, stream>>>(objectness, scores, hist);
  select_kernel<<<1, NIMG, 0, stream>>>(hist, prefix, krem, 24u);

  const unsigned shifts[3] = {16u, 8u, 0u};
  const unsigned hmasks[3] = {0xFF000000u, 0xFFFF0000u, 0xFFFFFF00u};
  for (int ps = 0; ps < 3; ++ps) {
    hist_kernel<<<NIMG * CHUNKS, 256, 0, stream>>>(scores, hist, prefix,
                                                   shifts[ps], hmasks[ps]);
    select_kernel<<<1, NIMG, 0, stream>>>(hist, prefix, krem, shifts[ps]);
  }

  compact_kernel<<<(total + 255) / 256, 256, 0, stream>>>(scores, prefix, ckey, cidx, ccnt);
  sort_kernel<<<NIMG, 512, 0, stream>>>(ckey, cidx, ccnt, top_idx, top_score);
  decode_kernel<<<(NIMG * TOPK + 255) / 256, 256, 0, stream>>>(box_reg, anchors,
                                                               top_idx, boxes);
  mask_kernel<<<(NIMG * TOPK * 32 + 255) / 256, 256, 0, stream>>>(boxes, supmask);
  nms_reduce_kernel<<<NIMG, 32, 0, stream>>>(supmask, ord, count);
  output_kernel<<<(NIMG * TOPK + 255) / 256, 256, 0, stream>>>(boxes, top_score,
                                                               ord, count, out);
}